// CategoricalDiffusion_38019050504196
// MI455X (gfx1250) — compile-verified
//
#include <hip/hip_runtime.h>
#include <cstddef>

typedef __attribute__((ext_vector_type(2))) float v2f;
typedef __attribute__((ext_vector_type(8))) float v8f;

#define KDIM   64
#define NBL    8192            /* B*L */
#define EPSF   1e-6f

// One merged LDS buffer; per-matrix bases are compile-time constants so every
// ds_load uses a single runtime base register + immediate offset.
#define PP_OFF 0               /* P[t-1] */
#define PC_OFF 5120            /* P[t]   */
#define AT_OFF 10240           /* A^T    */

__device__ __forceinline__ v8f wmma4(v2f a, v2f b, v8f c) {
  // D = A(16x4 f32) * B(4x16 f32) + C(16x16 f32)
  return __builtin_amdgcn_wmma_f32_16x16x4_f32(false, a, false, b, (short)0, c,
                                               false, false);
}

// Fragment-packed layout: element [k = 4*kk + 2*h + j][n] -> kk*320 + h*160
// + n*2 + j.  Row-group stride 160 dwords: the two half-wave b64 reads land
// 32 banks apart, so 32 lanes tile all 64 LDS banks conflict-free.
__device__ __forceinline__ int pidx(int kk, int h, int n) {
  return kk * 320 + h * 160 + n * 2;
}

__global__ void zero_kernel(float* out) {
  if (threadIdx.x == 0) out[0] = 0.0f;
}

// ---------------------------------------------------------------------------
// powers_kernel: block t computes A^t (binary exponentiation, LDS matmuls)
// ---------------------------------------------------------------------------
__device__ __forceinline__ void mm64(const float* __restrict__ X,
                                     const float* __restrict__ Y,
                                     float* __restrict__ Z, int tid) {
#pragma unroll
  for (int i0 = 0; i0 < 16; ++i0) {
    const int i = tid + i0 * 256;
    const int r = i >> 6, c = i & 63;
    float s = 0.0f;
#pragma unroll 8
    for (int k = 0; k < 64; ++k) s += X[r * 64 + k] * Y[k * 64 + c];
    Z[i] = s;
  }
}

__global__ __launch_bounds__(256) void powers_kernel(const float* __restrict__ A,
                                                     float* __restrict__ P) {
  __shared__ float R[4096], Bm[4096], Tm[4096];
  const int t = blockIdx.x;
  const int tid = threadIdx.x;
  for (int i = tid; i < 4096; i += 256) {
    R[i]  = ((i >> 6) == (i & 63)) ? 1.0f : 0.0f;
    Bm[i] = A[i];
  }
  __syncthreads();
  for (int bit = 0; bit < 6; ++bit) {        // t in [0,63]
    if (t & (1 << bit)) {                    // block-uniform branch
      mm64(R, Bm, Tm, tid);
      __syncthreads();
      for (int i = tid; i < 4096; i += 256) R[i] = Tm[i];
      __syncthreads();
    }
    if (bit < 5) {
      mm64(Bm, Bm, Tm, tid);
      __syncthreads();
      for (int i = tid; i < 4096; i += 256) Bm[i] = Tm[i];
      __syncthreads();
    }
  }
  float* dst = P + (size_t)t * 4096;
  for (int i = tid; i < 4096; i += 256) dst[i] = R[i];
}

// ---------------------------------------------------------------------------
// Main loss kernel: grid (64 site-groups, 63 t-values), 128 threads (4 waves).
// t = blockIdx.y + 1.  t==1 blocks also fold in the t=0/t=1 L_T terms + L_01.
//   phase A : accC (rp_cur) only, consumed by the denom pass
//   fused   : per column-block c (rolled loop): aP/aF/aN -> loss, discarded
// All LDS fragment reads: one base VGPR + immediate offsets.
// ---------------------------------------------------------------------------
__global__ __launch_bounds__(128, 4) void
diff_loss_kernel(const float* __restrict__ real,
                 const float* __restrict__ ns,
                 const float* __restrict__ yp,
                 const float* __restrict__ A,
                 const float* __restrict__ PxT,
                 const float* __restrict__ P,
                 float* __restrict__ out) {
  __shared__ float Bufs[3 * 5120];

  constexpr float W_LT  = 1.0f / (64.0f * 8192.0f);
  constexpr float W_01  = 1.0f / 8192.0f;
  constexpr float W_TM1 = 1.0f / (62.0f * 8192.0f);

  const int t    = blockIdx.y + 1;
  const int tid  = threadIdx.x;
  const int lane = tid & 31;
  const int wid  = tid >> 5;
  const int li   = lane & 15;
  const int h    = lane >> 4;

  // Stage B operands into packed LDS layout (coalesced global reads).
  {
    const float* gPc = P + (unsigned)t * (KDIM * KDIM);
    const float* gPp = P + (unsigned)(t - 1) * (KDIM * KDIM);
    for (int i = tid; i < KDIM * KDIM; i += 128) {
      const int k = i >> 6, n = i & 63;
      const int d = pidx(k >> 2, (k >> 1) & 1, n) + (k & 1);
      Bufs[PP_OFF + d] = gPp[i];
      Bufs[PC_OFF + d] = gPc[i];
      Bufs[AT_OFF + d] = A[n * KDIM + k];    // At[k][n] = A[n][k]
    }
  }
  __syncthreads();

  float acc = 0.0f;

  for (int it = 0; it < 2; ++it) {          // 2 site tiles per wave
    const int s0   = (blockIdx.x * 8 + wid * 2 + it) * 16;
    const int arow = s0 + li;               // A-fragment row for this lane

    if (t == 1) {
      // FC[1] = x0 @ A   (Pc == P[1] == A)
      v2f ax[16];
      const float* xb = ns + (unsigned)arow * KDIM + 2 * h;   // x0 = ns[0]
#pragma unroll
      for (int kk = 0; kk < 16; ++kk) ax[kk] = *(const v2f*)(xb + 4 * kk);
      float ltT = 0.0f, l01 = 0.0f;
#pragma unroll 1
      for (int c = 0; c < 4; ++c) {         // rolled: bound liveness
        const int bi0 = h * 160 + (c * 16 + li) * 2;
        v8f a = {};
#pragma unroll
        for (int kk = 0; kk < 16; ++kk) {
          v2f bc = *(const v2f*)&Bufs[PC_OFF + bi0 + kk * 320];
          a = wmma4(ax[kk], bc, a);
        }
        const float lpx = __logf(PxT[c * 16 + li] + EPSF);
#pragma unroll
        for (int r = 0; r < 8; ++r) {
          const unsigned o =
              (unsigned)(s0 + r + 8 * h) * KDIM + (unsigned)(c * 16 + li);
          const float n0 = ns[o] + EPSF;                            // t = 0
          const float n1 = ns[(unsigned)NBL * KDIM + o] + EPSF;     // t = 1
          ltT += n0 * (__logf(n0) - lpx);
          ltT += n1 * (__logf(n1) - lpx);
          l01 += a[r] *
                 __logf(__builtin_nontemporal_load(yp + (unsigned)NBL * KDIM + o));
        }
      }
      acc += ltT * W_LT - l01 * W_01;
    } else {
      const unsigned eb = ((unsigned)t * NBL + (unsigned)s0) * KDIM;

      // ---- Phase A: rp_cur = real @ P[t]  (for the denominator)
      v2f ar[16];
      {
        const float* rb = real + (unsigned)arow * KDIM + 2 * h;
#pragma unroll
        for (int kk = 0; kk < 16; ++kk) ar[kk] = *(const v2f*)(rb + 4 * kk);
      }
      v8f accC[4];
#pragma unroll
      for (int c = 0; c < 4; ++c) {
        const int bi0 = h * 160 + (c * 16 + li) * 2;
        v8f a = {};
#pragma unroll
        for (int kk = 0; kk < 16; ++kk) {
          v2f bc = *(const v2f*)&Bufs[PC_OFF + bi0 + kk * 320];
          a = wmma4(ar[kk], bc, a);
        }
        accC[c] = a;
      }

      // ---- Pass 1: L_T term + denom row-dot (m = r + 8h, n = c*16 + li).
      float dp[8] = {0, 0, 0, 0, 0, 0, 0, 0};
      float ltT = 0.0f;
#pragma unroll
      for (int c = 0; c < 4; ++c) {
        const float lpx = __logf(PxT[c * 16 + li] + EPSF);
#pragma unroll
        for (int r = 0; r < 8; ++r) {
          const unsigned o = eb + (unsigned)(r + 8 * h) * KDIM +
                             (unsigned)(c * 16 + li);
          const float xv = ns[o];
          const float xe = xv + EPSF;
          ltT += xe * (__logf(xe) - lpx);
          dp[r] += accC[c][r] * xv;
        }
      }
#pragma unroll
      for (int r = 0; r < 8; ++r) {   // reduce over 16-lane half (the n dim)
        dp[r] += __shfl_xor(dp[r], 1, 16);
        dp[r] += __shfl_xor(dp[r], 2, 16);
        dp[r] += __shfl_xor(dp[r], 4, 16);
        dp[r] += __shfl_xor(dp[r], 8, 16);
      }

      // ---- Load remaining A-fragments: x0 and xt
      v2f ax[16], an[16];
      {
        const float* xb = ns + (unsigned)arow * KDIM + 2 * h;              // x0
        const float* nb =
            ns + ((unsigned)t * NBL + (unsigned)arow) * KDIM + 2 * h;      // xt
#pragma unroll
        for (int kk = 0; kk < 16; ++kk) {
          ax[kk] = *(const v2f*)(xb + 4 * kk);
          an[kk] = *(const v2f*)(nb + 4 * kk);
        }
      }

      // ---- Fused: per column-block compute rp_prev / fc / numer1 and
      //      immediately fold into L_tm1.  Rolled loop bounds liveness;
      //      one LDS base register serves all 48 fragment loads.
      float ltm = 0.0f;
#pragma unroll 1
      for (int c = 0; c < 4; ++c) {
        const int bi0 = h * 160 + (c * 16 + li) * 2;
        v8f aP = {}, aF = {}, aN = {};
#pragma unroll
        for (int kk = 0; kk < 16; ++kk) {
          v2f bp = *(const v2f*)&Bufs[PP_OFF + bi0 + kk * 320];
          v2f bc = *(const v2f*)&Bufs[PC_OFF + bi0 + kk * 320];
          v2f bt = *(const v2f*)&Bufs[AT_OFF + bi0 + kk * 320];
          aP = wmma4(ar[kk], bp, aP);   // rp_prev = real @ P[t-1]
          aF = wmma4(ax[kk], bc, aF);   // fc      = x0   @ P[t]
          aN = wmma4(an[kk], bt, aN);   // numer1  = xt   @ A^T
        }
#pragma unroll
        for (int r = 0; r < 8; ++r) {
          const unsigned o = eb + (unsigned)(r + 8 * h) * KDIM +
                             (unsigned)(c * 16 + li);
          const float rev = aN[r] * aP[r] / dp[r];
          const float fcv = aF[r];
          const float px  = rev * fcv * __builtin_nontemporal_load(yp + o);
          ltm += fcv * rev * (__logf(rev + EPSF) - __logf(px + EPSF));
        }
      }
      acc += ltT * W_LT + ltm * W_TM1;
    }
  }

  // Wave reduce, then one atomic per wave.
  acc += __shfl_xor(acc, 1, 32);
  acc += __shfl_xor(acc, 2, 32);
  acc += __shfl_xor(acc, 4, 32);
  acc += __shfl_xor(acc, 8, 32);
  acc += __shfl_xor(acc, 16, 32);
  if (lane == 0) atomicAdd(out, acc);
}

// ---------------------------------------------------------------------------
extern "C" void kernel_launch(void* const* d_in, const int* in_sizes, int n_in,
                              void* d_out, int out_size, void* d_ws,
                              size_t ws_size, hipStream_t stream) {
  (void)in_sizes; (void)n_in; (void)out_size; (void)ws_size;
  const float* real = (const float*)d_in[0];   // (B,L,K)
  const float* ns   = (const float*)d_in[1];   // (T,B,L,K)
  const float* yp   = (const float*)d_in[2];   // (T,B,L,K)
  const float* A    = (const float*)d_in[3];   // (K,K)
  const float* PxT  = (const float*)d_in[4];   // (K)
  float* P   = (float*)d_ws;                   // 64 * 64*64 f32 = 1 MiB
  float* out = (float*)d_out;

  zero_kernel<<<1, 32, 0, stream>>>(out);
  powers_kernel<<<64, 256, 0, stream>>>(A, P);
  dim3 grid(64, 63);                            // x: site groups, y: t = 1..63
  diff_loss_kernel<<<grid, 128, 0, stream>>>(real, ns, yp, A, PxT, P, out);
}